// RPTNv1_72962904425170
// MI455X (gfx1250) — compile-verified
//
#include <hip/hip_runtime.h>

typedef __attribute__((ext_vector_type(2))) float v2f;
typedef __attribute__((ext_vector_type(8))) float v8f;

#define N_IMG   8
#define IN_CH   16
#define C1      2048
#define HW      3136
#define W_DIM   56
#define TILE    32
#define N_TILES (HW / TILE)   // 98, exact
#define C_OUT   2048
#define GRP     4
#define NTHREADS 512          // 16 wave32 waves; wave id == input channel

// One block = one image x 32 consecutive spatial positions (one aligned 128B
// line per output channel).
// Phase 1: grouped 3x3 conv as A[16x9pad12] x B[9pad12x16] via
//          v_wmma_f32_16x16x4_f32 (3 per tile, 2 N-tiles), bias folded into
//          the accumulator; all 2048 channels x 32 positions (256KB) in LDS.
// Phase 2: permutation gather + max-over-4 from LDS; each wave emits one
//          full 128B-aligned non-temporal store line.
__global__ __launch_bounds__(NTHREADS) void fused_dwconv_gather_max(
    const float* __restrict__ x,
    const float* __restrict__ W1,
    const float* __restrict__ b1,
    const int*   __restrict__ index,
    float* __restrict__ out)
{
    __shared__ float lds1[C1 * TILE];   // 256 KB of the WGP's 320 KB LDS

    const int wave = threadIdx.x >> 5;   // 0..15 == input channel
    const int lane = threadIdx.x & 31;
    const int half = lane >> 4;          // 0: lanes 0-15, 1: lanes 16-31
    const int sub  = lane & 15;
    const int n_img = blockIdx.x / N_TILES;
    const int tile  = blockIdx.x % N_TILES;
    const int p0 = tile * TILE;

    // ---------------- Phase 1: conv via WMMA ----------------
    {
        const int ic = wave;
        const float* xc = x + (size_t)(n_img * IN_CH + ic) * HW;

        // Two B tiles (N-tiles of 16 positions each); K = 9 zero-padded to 12,
        // split into 3 chunks of 4. 32-bit B 4x16 layout: lanes 0-15 hold
        // (K0,K1), lanes 16-31 hold (K2,K3); column N = sub.
        v2f Bm[2][3];
        #pragma unroll
        for (int nt = 0; nt < 2; ++nt) {
            const int p  = p0 + nt * 16 + sub;
            const int hh = p / W_DIM;
            const int ww = p % W_DIM;
            #pragma unroll
            for (int kc = 0; kc < 3; ++kc) {
                const int kr0 = kc * 4 + half * 2;
                float bv[2];
                #pragma unroll
                for (int j = 0; j < 2; ++j) {
                    const int k = kr0 + j;
                    float v = 0.0f;
                    if (k < 9) {
                        const int h2 = hh + (k / 3) - 1;
                        const int w2 = ww + (k % 3) - 1;
                        if (h2 >= 0 && h2 < W_DIM && w2 >= 0 && w2 < W_DIM)
                            v = xc[h2 * W_DIM + w2];
                    }
                    bv[j] = v;
                }
                Bm[nt][kc].x = bv[0];
                Bm[nt][kc].y = bv[1];
            }
        }

        #pragma unroll 1
        for (int ft = 0; ft < 8; ++ft) {
            const int cbase = ic * 128 + ft * 16;

            // A = 16 filters x K(pad 12). 32-bit A 16x4 layout: lane row
            // M = sub; lanes 0-15 hold (K0,K1), lanes 16-31 hold (K2,K3).
            const float* wf = W1 + (size_t)(cbase + sub) * 9;
            v2f A[3];
            #pragma unroll
            for (int kc = 0; kc < 3; ++kc) {
                const int kr0 = kc * 4 + half * 2;
                A[kc].x = (kr0     < 9) ? wf[kr0]     : 0.0f;
                A[kc].y = (kr0 + 1 < 9) ? wf[kr0 + 1] : 0.0f;
            }

            #pragma unroll
            for (int nt = 0; nt < 2; ++nt) {
                // C/D layout: VGPR r -> M = half*8 + r, N = sub. Fold bias in.
                v8f c;
                #pragma unroll
                for (int r = 0; r < 8; ++r)
                    c[r] = b1[cbase + half * 8 + r];

                c = __builtin_amdgcn_wmma_f32_16x16x4_f32(
                        false, A[0], false, Bm[nt][0], (short)0, c, false, false);
                c = __builtin_amdgcn_wmma_f32_16x16x4_f32(
                        false, A[1], false, Bm[nt][1], (short)0, c, false, false);
                c = __builtin_amdgcn_wmma_f32_16x16x4_f32(
                        false, A[2], false, Bm[nt][2], (short)0, c, false, false);

                #pragma unroll
                for (int r = 0; r < 8; ++r) {
                    const int m = half * 8 + r;
                    lds1[(cbase + m) * TILE + nt * 16 + sub] = c[r];
                }
            }
        }
    }

    __syncthreads();

    // ------- Phase 2: out[n, co, p] = max_g out1[n, index[co*4+g] % C1, p] ----
    // Each wave: one output channel x 32 positions = one aligned 128B NT store.
    for (int i = threadIdx.x; i < C_OUT * TILE; i += NTHREADS) {
        const int co = i >> 5;           // wave-uniform
        const int nn = i & 31;           // == lane
        float m = -3.402823466e38f;
        #pragma unroll
        for (int g = 0; g < GRP; ++g) {
            const int c1v = index[co * GRP + g] & (C1 - 1);
            m = fmaxf(m, lds1[c1v * TILE + nn]);
        }
        __builtin_nontemporal_store(
            m, &out[(size_t)(n_img * C_OUT + co) * HW + p0 + nn]);
    }
}

extern "C" void kernel_launch(void* const* d_in, const int* in_sizes, int n_in,
                              void* d_out, int out_size, void* d_ws, size_t ws_size,
                              hipStream_t stream) {
    const float* x   = (const float*)d_in[0];
    const float* W1  = (const float*)d_in[1];
    const float* b1  = (const float*)d_in[2];
    const int*   idx = (const int*)d_in[3];
    float* out = (float*)d_out;

    dim3 grid(N_IMG * N_TILES);   // 784 workgroups
    dim3 block(NTHREADS);         // 16 wave32 waves
    hipLaunchKernelGGL(fused_dwconv_gather_max, grid, block, 0, stream,
                       x, W1, b1, idx, out);
}